// Resample_30855045055084
// MI455X (gfx1250) — compile-verified
//
#include <hip/hip_runtime.h>
#include <math.h>

#ifndef M_PI
#define M_PI 3.14159265358979323846
#endif

#define T_IN  30000
#define N_OUT 24000
#define XLEN  108          // per-row staged input extent per block
#define XSTR  113          // LDS row stride (odd -> bank-conflict free)
#define WSTR  49           // W LDS row stride (odd)
#define WAVES 4
#define BLOCK_THREADS 128
#define OUT_PER_BLOCK 64   // 4 waves * 16 outputs

typedef __attribute__((ext_vector_type(2))) float v2f;
typedef __attribute__((ext_vector_type(8))) float v8f;

// Only check on the DEVICE pass -- the host pass never has amdgcn builtins.
#if defined(__HIP_DEVICE_COMPILE__) && defined(__gfx1250__)
#if !__has_builtin(__builtin_amdgcn_wmma_f32_16x16x4_f32)
#error "device pass: __builtin_amdgcn_wmma_f32_16x16x4_f32 not available on gfx1250"
#endif
#endif

__device__ inline double bessel_i0(double x) {
  double t = 0.25 * x * x, term = 1.0, s = 1.0;
  #pragma unroll 1
  for (int k = 1; k <= 40; ++k) { term *= t / ((double)k * (double)k); s += term; }
  return s;
}

__global__ __launch_bounds__(BLOCK_THREADS)
void resample_poly_wmma(const float* __restrict__ x, float* __restrict__ out) {
  __shared__ float  Xlds[16 * XSTR];
  __shared__ float  Wlds[16 * WSTR];
  __shared__ double h_lds[101];
  __shared__ double hsum_lds;
  __shared__ float  Dlds[WAVES][16][17];

  const int tid = threadIdx.x;

  // ---- prototype Kaiser-windowed sinc (fp64, matches scipy design) ----
  if (tid < 101) {
    int m = tid - 50;
    double g = (m == 0) ? 0.2 : sin(M_PI * 0.2 * (double)m) / (M_PI * (double)m);
    double r = (double)m / 50.0;
    double arg = 5.0 * sqrt(fmax(0.0, 1.0 - r * r));
    h_lds[tid] = g * bessel_i0(arg);   // 1/I0(beta) cancels in normalization
  }
  __syncthreads();
  if (tid == 0) {
    double s = 0.0;
    for (int k = 0; k < 101; ++k) s += h_lds[k];
    hsum_lds = s;
  }
  __syncthreads();

  // ---- universal weight tile: W[i][m] = H_ext[107 + 5i - 4m], i<16, m<48 ----
  for (int idx = tid; idx < 16 * 48; idx += BLOCK_THREADS) {
    int i = idx / 48, mm = idx - 48 * i;
    int t = 107 + 5 * i - 4 * mm;
    float wvv = 0.0f;
    if (t >= 5 && t <= 105)            // first 5 entries of padded H are zero
      wvv = (float)(h_lds[t - 5] * (4.0 / hsum_lds));
    Wlds[i * WSTR + mm] = wvv;
  }

  // ---- stage input: rows rowg..rowg+15, samples xbase..xbase+XLEN-1 ----
  const int rowg  = blockIdx.y * 16;
  const int n_t0  = blockIdx.x * OUT_PER_BLOCK;
  const int xbase = 5 * (n_t0 >> 2) - 13;
  for (int idx = tid; idx < 16 * XLEN; idx += BLOCK_THREADS) {
    int row = idx / XLEN, mm = idx - XLEN * row;
    int g = xbase + mm;
    float v = 0.0f;
    if (g >= 0 && g < T_IN) v = x[(size_t)(rowg + row) * T_IN + g];
    Xlds[row * XSTR + mm] = v;          // zero-fill: no NaN*0 into WMMA
  }
  __syncthreads();

  const int wv   = tid >> 5;
  const int lane = tid & 31;
  const int l15  = lane & 15;
  const int hi   = (lane >= 16) ? 2 : 0;   // K-half per ISA 16x4 A layout

  // A fragments (weights) once per wave, reused for all 12 K-steps
  v2f afrag[12];
  #pragma unroll
  for (int s = 0; s < 12; ++s) {
    int t = 4 * s + hi;
    v2f a; a.x = Wlds[l15 * WSTR + t]; a.y = Wlds[l15 * WSTR + t + 1];
    afrag[s] = a;
  }

  const int mbase = 20 * wv;               // xbase shift per 16-output tile
  v8f acc = {0.f, 0.f, 0.f, 0.f, 0.f, 0.f, 0.f, 0.f};
  #pragma unroll
  for (int s = 0; s < 12; ++s) {
    int k0 = mbase + 4 * s + hi;
    v2f b; b.x = Xlds[l15 * XSTR + k0]; b.y = Xlds[l15 * XSTR + k0 + 1];
    acc = __builtin_amdgcn_wmma_f32_16x16x4_f32(false, afrag[s], false, b,
                                                (short)0, acc, false, false);
  }

  // ---- transpose 16x16 tile through LDS for coalesced stores ----
  const int Mb = (lane < 16) ? 0 : 8;
  #pragma unroll
  for (int r = 0; r < 8; ++r) Dlds[wv][Mb + r][l15] = acc[r];
  __syncthreads();

  const int n_t = n_t0 + 16 * wv;
  #pragma unroll
  for (int q = 0; q < 8; ++q) {
    int e = lane + 32 * q;                 // 0..255
    int col = e & 15, row = e >> 4;        // 16 consecutive floats per row
    out[(size_t)(rowg + row) * N_OUT + n_t + col] = Dlds[wv][col][row];
  }
}

extern "C" void kernel_launch(void* const* d_in, const int* in_sizes, int n_in,
                              void* d_out, int out_size, void* d_ws, size_t ws_size,
                              hipStream_t stream) {
  (void)n_in; (void)d_ws; (void)ws_size; (void)out_size;
  const float* x = (const float*)d_in[0];
  float* out = (float*)d_out;
  const int rows = in_sizes[0] / T_IN;                 // 1536
  dim3 grid(N_OUT / OUT_PER_BLOCK, rows / 16);         // 375 x 96
  resample_poly_wmma<<<grid, BLOCK_THREADS, 0, stream>>>(x, out);
}